// RPN_SubqueryFeatureExtractor_77051713290505
// MI455X (gfx1250) — compile-verified
//
#include <hip/hip_runtime.h>

typedef float v2f __attribute__((ext_vector_type(2)));
typedef float v8f __attribute__((ext_vector_type(8)));

#define B_    8
#define N_    100
#define C_    256
#define G_    4
#define P_    32
#define POOL_ 9
#define CG_   64   // C_/G_

// ---------------------------------------------------------------------------
// fp32 WMMA GEMM:  D = A(MxK) * Wt(NoutxK)^T + bias  ->  D(M x Nout)
// One wave computes one 16x16 output tile using V_WMMA_F32_16X16X4_F32.
// A-frag layout (16x4 f32): lanes 0-15: row=lane, (K=k,k+1); lanes 16-31:
// row=lane-16, (K=k+2,k+3).  B-frag (4x16): lanes 0-15: col=lane, (K=k,k+1);
// lanes 16-31: col=lane-16, (K=k+2,k+3).  Both are contiguous float2 loads
// because B columns are rows of Wt.
// C/D (16x16 f32, 8 VGPRs): lane<16 -> (M=r, N=lane); lane>=16 -> (M=r+8).
// ---------------------------------------------------------------------------
__global__ __launch_bounds__(256) void gemm_wmma_f32(
    const float* __restrict__ A, const float* __restrict__ Wt,
    const float* __restrict__ bias, float* __restrict__ D,
    int M, int Nout, int K)
{
    const int wid  = (int)((blockIdx.x * blockDim.x + threadIdx.x) >> 5);
    const int lane = threadIdx.x & 31;
    const int ntiles = Nout >> 4;
    const int totalWaves = (M >> 4) * ntiles;
    if (wid >= totalWaves) return;

    const int mt   = wid / ntiles;
    const int nt   = wid - mt * ntiles;
    const int half = lane >> 4;   // 0: lanes 0-15, 1: lanes 16-31
    const int l16  = lane & 15;

    const float* Ar = A  + (size_t)(mt * 16 + l16) * K + half * 2;
    const float* Wr = Wt + (size_t)(nt * 16 + l16) * K + half * 2;

    v8f acc = {};
    for (int k = 0; k < K; k += 4) {
        v2f a = *(const v2f*)(Ar + k);
        v2f b = *(const v2f*)(Wr + k);
        acc = __builtin_amdgcn_wmma_f32_16x16x4_f32(
            /*neg_a=*/false, a, /*neg_b=*/false, b,
            /*c_mod=*/(short)0, acc, /*reuse_a=*/false, /*reuse_b=*/false);
    }

    const int   ocol = nt * 16 + l16;
    const float bv   = bias[ocol];
    const int   orow = mt * 16 + half * 8;
    float* Dp = D + (size_t)orow * Nout + ocol;
#pragma unroll
    for (int r = 0; r < 8; ++r)
        Dp[(size_t)r * Nout] = acc[r] + bv;
}

// ---------------------------------------------------------------------------
// Sampler: one wave per (b, n, g, p, pool) point. lane = channel c, and the
// same lane also handles channel c+32 (Cg=64). Per-point scalar math is
// redundant across lanes (cheap VALU); the gathers dominate.
// ---------------------------------------------------------------------------
__global__ __launch_bounds__(256) void sample_kernel(
    const float* __restrict__ f0, const float* __restrict__ f1,
    const float* __restrict__ f2, const float* __restrict__ f3,
    const float* __restrict__ qb, const float* __restrict__ D1,
    const float* __restrict__ D2, float* __restrict__ out)
{
    const int S    = B_ * N_ * G_ * P_ * POOL_;     // 921600 sample points
    const int wid  = (int)((blockIdx.x * blockDim.x + threadIdx.x) >> 5);
    const int lane = threadIdx.x & 31;
    if (wid >= S) return;

    int k = wid % POOL_;
    int t = wid / POOL_;
    int p = t & (P_ - 1); t >>= 5;
    int g = t & (G_ - 1); t >>= 2;
    int n = t % N_;
    int b = t / N_;

    // query box
    const float* qbp = qb + (size_t)(b * N_ + n) * 4;
    const float bx = qbp[0], by = qbp[1], bz = qbp[2], br = qbp[3];

    // d = qc@w_sub.T + b_sub   (4th component replaced by 1.0 in reference)
    const float* dp = D1 + (size_t)(b * N_ + n) * (G_ * P_ * 4) + (g * P_ + p) * 4;
    const float dx = dp[0], dy = dp[1], dz = dp[2];

    // kpts = qc@w_pts.T + b_pts
    const float* kp = D2 + (size_t)(b * N_ + n) * (G_ * P_ * POOL_ * 2)
                         + ((g * P_ + p) * POOL_ + k) * 2;
    const float kx = kp[0], ky = kp[1];

    // sub-box decode
    const float ow = exp2f(bz - 0.5f * br);
    const float oh = exp2f(bz + 0.5f * br);
    const float cx = bx + dx * ow;
    const float cy = by + dy * oh;
    const float nz = bz + dz;
    const float nr = br + 1.0f;

    // softmax over levels: wz_l ∝ exp(-((nz-3)-l)^2 / 2)
    const float u  = nz - 3.0f;
    const float u1 = u - 1.0f, u2 = u - 2.0f, u3 = u - 3.0f;
    float e0 = expf(-0.5f * u  * u);
    float e1 = expf(-0.5f * u1 * u1);
    float e2 = expf(-0.5f * u2 * u2);
    float e3 = expf(-0.5f * u3 * u3);
    const float inv = 1.0f / (e0 + e1 + e2 + e3);
    const float wz[4] = { e0 * inv, e1 * inv, e2 * inv, e3 * inv };

    // sample position
    const float sw = exp2f(nz - 0.5f * nr);
    const float sh = exp2f(nz + 0.5f * nr);
    const float gx = cx + kx * sw;
    const float gy = cy + ky * sh;

    const float* feats[4] = { f0, f1, f2, f3 };
    const int   Wl[4] = { 256, 128, 64, 32 };
    const int   Hl[4] = { 192,  96, 48, 24 };
    const float is[4] = { 0.25f, 0.125f, 0.0625f, 0.03125f };  // 1/stride

    float acc0 = 0.0f, acc1 = 0.0f;
#pragma unroll
    for (int lv = 0; lv < 4; ++lv) {
        const int W = Wl[lv], H = Hl[lv];
        // all levels map to the same 1024x768 extent; x = gx/stride - 0.5
        const float xx = gx * is[lv] - 0.5f;
        const float yy = gy * is[lv] - 0.5f;
        const float fx = floorf(xx), fy = floorf(yy);
        const float wx1 = xx - fx, wy1 = yy - fy;
        const float wx0 = 1.0f - wx1, wy0 = 1.0f - wy1;
        const int ix0 = (int)fx, iy0 = (int)fy;
        const int ix1 = ix0 + 1, iy1 = iy0 + 1;
        // zero-padding: fold validity masks into the bilinear weights
        const float mx0 = (ix0 >= 0 && ix0 < W) ? 1.0f : 0.0f;
        const float mx1 = (ix1 >= 0 && ix1 < W) ? 1.0f : 0.0f;
        const float my0 = (iy0 >= 0 && iy0 < H) ? 1.0f : 0.0f;
        const float my1 = (iy1 >= 0 && iy1 < H) ? 1.0f : 0.0f;
        const int cx0 = min(max(ix0, 0), W - 1);
        const int cx1 = min(max(ix1, 0), W - 1);
        const int cy0 = min(max(iy0, 0), H - 1);
        const int cy1 = min(max(iy1, 0), H - 1);
        const float w00 = wx0 * wy0 * mx0 * my0;
        const float w01 = wx1 * wy0 * mx1 * my0;
        const float w10 = wx0 * wy1 * mx0 * my1;
        const float w11 = wx1 * wy1 * mx1 * my1;

        const size_t hw = (size_t)H * W;
        const float* base = feats[lv] + ((size_t)(b * C_ + g * CG_ + lane)) * hw;
        const float* ch1  = base + (size_t)32 * hw;
        const int i00 = cy0 * W + cx0;
        const int i01 = cy0 * W + cx1;
        const int i10 = cy1 * W + cx0;
        const int i11 = cy1 * W + cx1;

        const float s0 = base[i00] * w00 + base[i01] * w01
                       + base[i10] * w10 + base[i11] * w11;
        const float s1 = ch1[i00]  * w00 + ch1[i01]  * w01
                       + ch1[i10]  * w10 + ch1[i11]  * w11;
        acc0 += s0 * wz[lv];
        acc1 += s1 * wz[lv];
    }

    // out: (B, N, G, P*POOL, Cg)
    const size_t ob = (((size_t)(b * N_ + n) * G_ + g) * (P_ * POOL_)
                       + (p * POOL_ + k)) * (size_t)CG_;
    out[ob + lane]      = acc0;
    out[ob + lane + 32] = acc1;
}

// ---------------------------------------------------------------------------
extern "C" void kernel_launch(void* const* d_in, const int* in_sizes, int n_in,
                              void* d_out, int out_size, void* d_ws, size_t ws_size,
                              hipStream_t stream)
{
    const float* f0    = (const float*)d_in[0];
    const float* f1    = (const float*)d_in[1];
    const float* f2    = (const float*)d_in[2];
    const float* f3    = (const float*)d_in[3];
    const float* qc    = (const float*)d_in[4];
    const float* qb    = (const float*)d_in[5];
    const float* w_sub = (const float*)d_in[6];
    const float* b_sub = (const float*)d_in[7];
    const float* w_pts = (const float*)d_in[8];
    const float* b_pts = (const float*)d_in[9];
    float* out = (float*)d_out;

    const int M  = B_ * N_;           // 800
    const int N1 = 4 * G_ * P_;       // 512
    const int N2 = 2 * G_ * P_ * POOL_; // 2304

    float* D1 = (float*)d_ws;                       // 800*512  floats
    float* D2 = D1 + (size_t)M * N1;                // 800*2304 floats

    // GEMM 1: d = qc @ w_sub.T + b_sub
    {
        const int waves  = (M / 16) * (N1 / 16);    // 1600
        const int blocks = (waves + 7) / 8;
        gemm_wmma_f32<<<blocks, 256, 0, stream>>>(qc, w_sub, b_sub, D1, M, N1, C_);
    }
    // GEMM 2: kpts = qc @ w_pts.T + b_pts
    {
        const int waves  = (M / 16) * (N2 / 16);    // 7200
        const int blocks = (waves + 7) / 8;
        gemm_wmma_f32<<<blocks, 256, 0, stream>>>(qc, w_pts, b_pts, D2, M, N2, C_);
    }
    // Sampler: one wave per (b,n,g,p,pool) point, 8 waves per block
    {
        const int S      = B_ * N_ * G_ * P_ * POOL_;  // 921600
        const int blocks = (S + 7) / 8;
        sample_kernel<<<blocks, 256, 0, stream>>>(f0, f1, f2, f3, qb, D1, D2, out);
    }
}